// SparseMoEEncoder_47966194761916
// MI455X (gfx1250) — compile-verified
//
#include <hip/hip_runtime.h>
#include <hip/hip_bf16.h>

typedef __attribute__((ext_vector_type(2))) float v2f;
typedef __attribute__((ext_vector_type(8))) float v8f;

// Problem constants (fixed by the reference)
#define BATCH 16
#define CC    3
#define HH    512
#define WW    512
#define PP    32
#define HP    16      // 512/32
#define WP    16
#define NP    (HP*WP)         // 256 patches per image
#define BN    (BATCH*NP)      // 4096 patches total
#define KDIM  (CC*PP*PP)      // 3072
#define DDIM  128
#define EE    8

// ---------------------------------------------------------------------------
// Kernel 1: gating. One block = 16 patches. 256 threads = 8 waves; wave w
// owns output columns [16w, 16w+16). WMMA f32 16x16x4 over K=3072.
// ---------------------------------------------------------------------------
__global__ __launch_bounds__(256) void moe_gate_kernel(
    const float* __restrict__ x,
    const float* __restrict__ w1, const float* __restrict__ b1,
    const float* __restrict__ w2, const float* __restrict__ b2,
    int* __restrict__ topk_idx, float* __restrict__ topk_w)
{
    __shared__ float As[16][260];   // 16 patches x 256-K chunk, stride 260 (mod64==4) -> no bank conflicts
    __shared__ float Hs[16][DDIM];  // hidden activations
    __shared__ float Ls[16][EE];    // logits

    const int tile  = blockIdx.x;          // group of 16 patches
    const int t     = threadIdx.x;
    const int wave  = t >> 5;
    const int lane  = t & 31;
    const int ln16  = lane & 15;
    const int khalf = lane >> 4;           // 0: K 0..1, 1: K 2..3
    const int ncol  = wave * 16 + ln16;    // output column for this lane

    v8f acc = {0.f,0.f,0.f,0.f,0.f,0.f,0.f,0.f};

    for (int kc = 0; kc < KDIM; kc += 256) {
        // Cooperative stage of A chunk (coalesced over j within patch rows)
        for (int idx = t; idx < 16 * 256; idx += 256) {
            int p  = idx >> 8;
            int kk = idx & 255;
            int k  = kc + kk;
            int c  = k >> 10;              // k / 1024
            int r  = k & 1023;
            int i  = r >> 5;
            int j  = r & 31;
            int pid = tile * 16 + p;
            int b   = pid >> 8;            // / NP
            int n   = pid & 255;
            int ph  = n >> 4, pw = n & 15;
            As[p][kk] = x[(((size_t)(b * CC + c) * HH) + (ph * PP + i)) * WW + (pw * PP + j)];
        }
        __syncthreads();

        // 64 WMMA steps over this K chunk
        for (int ks = 0; ks < 256; ks += 4) {
            int ka = ks + 2 * khalf;
            v2f a, bf;
            a.x = As[ln16][ka];
            a.y = As[ln16][ka + 1];
            int kg = kc + ka;
            bf.x = w1[(size_t)kg * DDIM + ncol];
            bf.y = w1[(size_t)(kg + 1) * DDIM + ncol];
            acc = __builtin_amdgcn_wmma_f32_16x16x4_f32(false, a, false, bf,
                                                        (short)0, acc, false, false);
        }
        __syncthreads();
    }

    // relu(acc + b1) -> Hs.  VGPR r holds M = r (lanes 0-15) / r+8 (lanes 16-31)
    float bias1 = b1[ncol];
#pragma unroll
    for (int r = 0; r < 8; ++r) {
        int m = r + 8 * khalf;
        float h = acc[r] + bias1;
        Hs[m][ncol] = h > 0.f ? h : 0.f;
    }
    __syncthreads();

    // logits: 16 patches x 8 experts
    if (t < 16 * EE) {
        int p = t >> 3, e = t & 7;
        float s = b2[e];
#pragma unroll 8
        for (int d = 0; d < DDIM; ++d) s += Hs[p][d] * w2[d * EE + e];
        Ls[p][e] = s;
    }
    __syncthreads();

    // top-2 + softmax (matches lax.top_k tie-breaking: earlier index wins)
    if (t < 16) {
        float v0 = -1e30f, v1 = -1e30f;
        int   i0 = 0, i1 = 0;
#pragma unroll
        for (int e = 0; e < EE; ++e) {
            float v = Ls[t][e];
            if (v > v0)      { v1 = v0; i1 = i0; v0 = v; i0 = e; }
            else if (v > v1) { v1 = v; i1 = e; }
        }
        float e0 = __expf(0.f);            // v0 - v0
        float e1 = __expf(v1 - v0);
        float inv = 1.f / (e0 + e1);
        int pid = tile * 16 + t;
        topk_idx[pid * 2 + 0] = i0;
        topk_idx[pid * 2 + 1] = i1;
        topk_w[pid * 2 + 0]   = e0 * inv;
        topk_w[pid * 2 + 1]   = e1 * inv;
    }
}

// ---------------------------------------------------------------------------
// Kernel 2: one block per patch; runs its two selected experts sequentially.
// dw1 + fused(pw1,dw2) on VALU, pw2 (the 2.1 MFLOP GEMM) on WMMA f32.
// ---------------------------------------------------------------------------
__global__ __launch_bounds__(256) void moe_expert_kernel(
    const float* __restrict__ x,
    const int*   __restrict__ topk_idx, const float* __restrict__ topk_w,
    const float* __restrict__ dw1w, const float* __restrict__ dw1b,
    const float* __restrict__ pw1w, const float* __restrict__ pw1b,
    const float* __restrict__ dw2w, const float* __restrict__ dw2b,
    const float* __restrict__ pw2w, const float* __restrict__ pw2b,
    float* __restrict__ out)
{
    __shared__ float patch[CC][32][33];   // 12.7 KB
    __shared__ float y1[CC][16][17];      //  3.3 KB
    __shared__ float zs[64][132];         // 33.8 KB, stride 132 (mod64==4) -> conflict-free float2 reads
    __shared__ float accum[DDIM];

    const int pid  = blockIdx.x;
    const int b    = pid >> 8;
    const int n    = pid & 255;
    const int ph   = n >> 4, pw = n & 15;
    const int t    = threadIdx.x;
    const int wave = t >> 5;
    const int lane = t & 31;
    const int ln16 = lane & 15;
    const int khalf = lane >> 4;
    const int ncol = wave * 16 + ln16;    // dout column for this lane

    // Stage the 3x32x32 patch
    for (int idx = t; idx < CC * 32 * 32; idx += 256) {
        int c = idx >> 10;
        int r = idx & 1023;
        int i = r >> 5, j = r & 31;
        patch[c][i][j] = x[(((size_t)(b * CC + c) * HH) + ph * PP + i) * WW + pw * PP + j];
    }
    if (t < DDIM) accum[t] = 0.f;
    __syncthreads();

    for (int kk = 0; kk < 2; ++kk) {
        const int   e  = topk_idx[pid * 2 + kk];
        const float gw = topk_w[pid * 2 + kk];

        // ---- dw1: depthwise 3x3 stride2 pad1 on 3 channels -> y1[3][16][16]
        {
            int oy = t >> 4, ox = t & 15;
#pragma unroll
            for (int c = 0; c < CC; ++c) {
                float s = dw1b[e * CC + c];
#pragma unroll
                for (int di = 0; di < 3; ++di) {
                    int iy = 2 * oy - 1 + di;
                    if (iy < 0 || iy >= 32) continue;
#pragma unroll
                    for (int dj = 0; dj < 3; ++dj) {
                        int ix = 2 * ox - 1 + dj;
                        if (ix < 0 || ix >= 32) continue;
                        s += patch[c][iy][ix] * dw1w[(e * CC + c) * 9 + di * 3 + dj];
                    }
                }
                y1[c][oy][ox] = s > 0.f ? s : 0.f;
            }
        }
        __syncthreads();

        // ---- fused pw1 (3->128, relu) + dw2 (depthwise 3x3 stride2 pad1, relu)
        //      z[d][q] for d in [0,128), q in [0,64)   (pw1 recomputed ~2.25x, saves 128KB LDS)
        for (int idx = t; idx < DDIM * 64; idx += 256) {
            int d = idx >> 6, q = idx & 63;
            int qy = q >> 3, qx = q & 7;
            float wc0 = pw1w[(e * DDIM + d) * 3 + 0];
            float wc1 = pw1w[(e * DDIM + d) * 3 + 1];
            float wc2 = pw1w[(e * DDIM + d) * 3 + 2];
            float pb  = pw1b[e * DDIM + d];
            float s   = dw2b[e * DDIM + d];
#pragma unroll
            for (int di = 0; di < 3; ++di) {
                int iy = 2 * qy - 1 + di;
                if (iy < 0 || iy >= 16) continue;
#pragma unroll
                for (int dj = 0; dj < 3; ++dj) {
                    int ix = 2 * qx - 1 + dj;
                    if (ix < 0 || ix >= 16) continue;
                    float v = pb + wc0 * y1[0][iy][ix] + wc1 * y1[1][iy][ix] + wc2 * y1[2][iy][ix];
                    v = v > 0.f ? v : 0.f;
                    s += v * dw2w[(e * DDIM + d) * 9 + di * 3 + dj];
                }
            }
            zs[q][d] = s > 0.f ? s : 0.f;
        }
        __syncthreads();

        // ---- pw2 GEMM via WMMA: M=64 (q), K=128 (din), this wave's N-tile = 16 douts
        v8f c0 = {0,0,0,0,0,0,0,0}, c1 = {0,0,0,0,0,0,0,0};
        v8f c2 = {0,0,0,0,0,0,0,0}, c3 = {0,0,0,0,0,0,0,0};
        const float* wrow = pw2w + (size_t)(e * DDIM + ncol) * DDIM; // [dout][din]
        for (int k = 0; k < DDIM; k += 4) {
            int ka = k + 2 * khalf;
            v2f bf;
            bf.x = wrow[ka];
            bf.y = wrow[ka + 1];
            v2f a0, a1, a2, a3;
            a0.x = zs[ 0 + ln16][ka];  a0.y = zs[ 0 + ln16][ka + 1];
            a1.x = zs[16 + ln16][ka];  a1.y = zs[16 + ln16][ka + 1];
            a2.x = zs[32 + ln16][ka];  a2.y = zs[32 + ln16][ka + 1];
            a3.x = zs[48 + ln16][ka];  a3.y = zs[48 + ln16][ka + 1];
            c0 = __builtin_amdgcn_wmma_f32_16x16x4_f32(false, a0, false, bf, (short)0, c0, false, false);
            c1 = __builtin_amdgcn_wmma_f32_16x16x4_f32(false, a1, false, bf, (short)0, c1, false, false);
            c2 = __builtin_amdgcn_wmma_f32_16x16x4_f32(false, a2, false, bf, (short)0, c2, false, false);
            c3 = __builtin_amdgcn_wmma_f32_16x16x4_f32(false, a3, false, bf, (short)0, c3, false, false);
        }

        // bias + relu + mean over the 64 positions, folded into the reduction
        float bias = pw2b[e * DDIM + ncol];
        float part = 0.f;
#pragma unroll
        for (int r = 0; r < 8; ++r) {
            float v;
            v = c0[r] + bias; part += v > 0.f ? v : 0.f;
            v = c1[r] + bias; part += v > 0.f ? v : 0.f;
            v = c2[r] + bias; part += v > 0.f ? v : 0.f;
            v = c3[r] + bias; part += v > 0.f ? v : 0.f;
        }
        part += __shfl_xor(part, 16, 32);   // combine the two M-halves
        if (lane < 16) accum[ncol] += gw * (part * (1.f / 64.f));
        __syncthreads();
    }

    // out layout: [B, D, hp, wp]
    if (t < DDIM) {
        out[(((size_t)b * DDIM + t) * HP + ph) * WP + pw] = accum[t];
    }
}

// ---------------------------------------------------------------------------
extern "C" void kernel_launch(void* const* d_in, const int* in_sizes, int n_in,
                              void* d_out, int out_size, void* d_ws, size_t ws_size,
                              hipStream_t stream) {
    const float* x       = (const float*)d_in[0];
    const float* gate_w1 = (const float*)d_in[1];
    const float* gate_b1 = (const float*)d_in[2];
    const float* gate_w2 = (const float*)d_in[3];
    const float* gate_b2 = (const float*)d_in[4];
    const float* dw1_w   = (const float*)d_in[5];
    const float* dw1_b   = (const float*)d_in[6];
    const float* pw1_w   = (const float*)d_in[7];
    const float* pw1_b   = (const float*)d_in[8];
    const float* dw2_w   = (const float*)d_in[9];
    const float* dw2_b   = (const float*)d_in[10];
    const float* pw2_w   = (const float*)d_in[11];
    const float* pw2_b   = (const float*)d_in[12];
    float* out = (float*)d_out;

    int*   topk_idx = (int*)d_ws;
    float* topk_w   = (float*)((char*)d_ws + (size_t)BN * 2 * sizeof(int));

    moe_gate_kernel<<<BN / 16, 256, 0, stream>>>(
        x, gate_w1, gate_b1, gate_w2, gate_b2, topk_idx, topk_w);

    moe_expert_kernel<<<BN, 256, 0, stream>>>(
        x, topk_idx, topk_w,
        dw1_w, dw1_b, pw1_w, pw1_b, dw2_w, dw2_b, pw2_w, pw2_b, out);
}